// MarchingSquares_63058709840404
// MI455X (gfx1250) — compile-verified
//
#include <hip/hip_runtime.h>
#include <cstdint>
#include <cstddef>

// ---------------------------------------------------------------------------
// Marching squares on MI455X (gfx1250).
// Memory-bound: ~370 MB traffic -> ~16us floor @ 23.3 TB/s.
// CDNA5 paths: global_load_async_to_lds_b32 (ASYNCcnt) staging,
// s_wait_asynccnt, LDS stencil reads, non-temporal b128 stores.
// Compute trimmed to ~130 VALU/cell (4 shared-inverse edge crossings)
// so VALU time (~14us) stays under the HBM store floor.
// ---------------------------------------------------------------------------

typedef float v4f __attribute__((ext_vector_type(4)));
typedef float v2f __attribute__((ext_vector_type(2)));

constexpr int B_  = 8;
constexpr int H_  = 1024;
constexpr int W_  = 1024;
constexpr int HC_ = H_ + 1;   // 1025
constexpr int WC_ = W_ + 1;   // 1025
constexpr int TILE_W = 64;
constexpr int TILE_H = 16;
constexpr long long EDGES_TOTAL = (long long)B_ * HC_ * WC_ * 8; // 67,240,000

// Edge-id table. Edges: S=0 (SW-SE), E=1 (SE-NE), N=2 (NE-NW), W=3 (NW-SW).
// The interpolated crossing is bit-exact symmetric in corner order, so each
// endpoint only needs its edge id.
// bits [1:0]=seg0.ep0 edge, [3:2]=seg0.ep1, [5:4]=seg1.ep0, [7:6]=seg1.ep1
// bit 8 = seg0 valid, bit 9 = seg1 valid
__constant__ unsigned kEdgeTab[16] = {
    0x000u, // 0
    0x103u, // 1:  (W,S)
    0x104u, // 2:  (S,E)
    0x107u, // 3:  (W,E)
    0x109u, // 4:  (E,N)
    0x393u, // 5:  (W,S) + (E,N)
    0x108u, // 6:  (S,N)
    0x10Bu, // 7:  (W,N)
    0x10Eu, // 8:  (N,W)
    0x102u, // 9:  (N,S)
    0x3E4u, // 10: (S,E) + (N,W)
    0x106u, // 11: (N,E)
    0x10Du, // 12: (E,W)
    0x101u, // 13: (E,S)
    0x10Cu, // 14: (S,W)
    0x000u  // 15
};

// ---- CDNA5 async global->LDS copy (ASYNCcnt path) -------------------------
__device__ __forceinline__ void async_copy_b32(const float* g, float* l) {
#if __has_builtin(__builtin_amdgcn_global_load_async_to_lds_b32)
    __builtin_amdgcn_global_load_async_to_lds_b32(
        (__attribute__((address_space(1))) int*)(uintptr_t)g,
        (__attribute__((address_space(3))) int*)l,
        0, 0);
#else
    unsigned lds_addr = (unsigned)(uintptr_t)l;  // low 32 bits == LDS byte addr
    asm volatile("global_load_async_to_lds_b32 %0, %1, off"
                 :
                 : "v"(lds_addr), "v"((unsigned long long)(uintptr_t)g)
                 : "memory");
#endif
}

__device__ __forceinline__ void wait_async0() {
#if __has_builtin(__builtin_amdgcn_s_wait_asynccnt)
    __builtin_amdgcn_s_wait_asynccnt(0);
#else
    asm volatile("s_wait_asynccnt 0" ::: "memory");
#endif
}

__device__ __forceinline__ float sel4(float a, float b, float c, float d,
                                      unsigned i) {
    return (i == 0u) ? a : (i == 1u) ? b : (i == 2u) ? c : d;
}

__global__ __launch_bounds__(256)
void MarchingSquares_63058709840404_kernel(const float* __restrict__ coarse,
                                           float* __restrict__ out) {
    __shared__ float tile[TILE_H + 1][TILE_W + 2];  // 17 x 66 f32 = 4.4 KB

    const int j0 = blockIdx.x * TILE_W;
    const int i0 = blockIdx.y * TILE_H;
    const int b  = blockIdx.z;
    const int tx = threadIdx.x;                 // 0..63
    const int ty = threadIdx.y;                 // 0..3
    const int tid = ty * 64 + tx;

    const float* src = coarse + (size_t)b * (H_ * W_);

    // Stage padded (TILE_H+1)x(TILE_W+1) window, edge-clamped, via async DMA.
    constexpr int NSTAGE = (TILE_H + 1) * (TILE_W + 1);  // 1105
    for (int f = tid; f < NSTAGE; f += 256) {
        int r = f / (TILE_W + 1);
        int c = f - r * (TILE_W + 1);
        int gr = i0 + r - 1;
        gr = gr < 0 ? 0 : (gr > H_ - 1 ? H_ - 1 : gr);
        int gc = j0 + c - 1;
        gc = gc < 0 ? 0 : (gc > W_ - 1 ? W_ - 1 : gc);
        async_copy_b32(src + (size_t)gr * W_ + gc, &tile[r][c]);
    }
    wait_async0();
    __syncthreads();

    float* __restrict__ edges = out;
    float* __restrict__ validp = out + EDGES_TOTAL;

#pragma unroll
    for (int k = 0; k < 4; ++k) {
        const int ry = ty + 4 * k;             // 0..15
        const int i  = i0 + ry;
        const int j  = j0 + tx;
        if (i >= HC_ || j >= WC_) continue;

        const float nw = tile[ry][tx];
        const float ne = tile[ry][tx + 1];
        const float sw = tile[ry + 1][tx];
        const float se = tile[ry + 1][tx + 1];

        const unsigned cfg = (sw > 0.f ? 1u : 0u) | (se > 0.f ? 2u : 0u) |
                             (ne > 0.f ? 4u : 0u) | (nw > 0.f ? 8u : 0u);
        const unsigned t = kEdgeTab[cfg];
        const bool vld0 = (t >> 8) & 1u;
        const bool vld1 = (t >> 9) & 1u;

        const float fi = (float)i;
        const float fj = (float)j;
        const float ys = fi + 0.5f, yn = fi - 0.5f;   // south / north row y
        const float xw = fj - 0.5f, xe = fj + 0.5f;   // west / east col x

        // Four canonical edge crossings; 1 IEEE division each.
        // Uncrossed edges may yield NaN/Inf but are never selected.
        const float invS = 1.0f / (se - sw);
        const float yS = (se * ys - sw * ys) * invS;
        const float xS = (se * xe - sw * xw) * invS;

        const float invE = 1.0f / (ne - se);
        const float yE = (ne * yn - se * ys) * invE;
        const float xE = (ne * xe - se * xe) * invE;

        const float invN = 1.0f / (nw - ne);
        const float yN = (nw * yn - ne * yn) * invN;
        const float xN = (nw * xw - ne * xe) * invN;

        const float invW = 1.0f / (sw - nw);
        const float yW = (sw * ys - nw * yn) * invW;
        const float xW = (sw * xw - nw * xw) * invW;

        const unsigned e00 = t & 3u;
        const unsigned e01 = (t >> 2) & 3u;
        const unsigned e10 = (t >> 4) & 3u;
        const unsigned e11 = (t >> 6) & 3u;

        v4f e0, e1;
        e0.x = vld0 ? sel4(yS, yE, yN, yW, e00) : 0.f;
        e0.y = vld0 ? sel4(xS, xE, xN, xW, e00) : 0.f;
        e0.z = vld0 ? sel4(yS, yE, yN, yW, e01) : 0.f;
        e0.w = vld0 ? sel4(xS, xE, xN, xW, e01) : 0.f;
        e1.x = vld1 ? sel4(yS, yE, yN, yW, e10) : 0.f;
        e1.y = vld1 ? sel4(xS, xE, xN, xW, e10) : 0.f;
        e1.z = vld1 ? sel4(yS, yE, yN, yW, e11) : 0.f;
        e1.w = vld1 ? sel4(xS, xE, xN, xW, e11) : 0.f;

        const size_t cell = ((size_t)b * HC_ + i) * WC_ + j;
        // Streaming output: non-temporal so the 336MB write stream does not
        // evict the (reused, 33.5MB) input from L2.
        __builtin_nontemporal_store(e0, (v4f*)(edges + cell * 8));
        __builtin_nontemporal_store(e1, (v4f*)(edges + cell * 8 + 4));
        v2f v2;
        v2.x = vld0 ? 1.f : 0.f;
        v2.y = vld1 ? 1.f : 0.f;
        __builtin_nontemporal_store(v2, (v2f*)(validp + cell * 2));
    }
}

extern "C" void kernel_launch(void* const* d_in, const int* in_sizes, int n_in,
                              void* d_out, int out_size, void* d_ws,
                              size_t ws_size, hipStream_t stream) {
    (void)in_sizes; (void)n_in; (void)d_ws; (void)ws_size; (void)out_size;
    const float* coarse = (const float*)d_in[0];
    float* out = (float*)d_out;

    dim3 block(64, 4, 1);  // 256 threads = 8 wave32
    dim3 grid((WC_ + TILE_W - 1) / TILE_W,   // 17
              (HC_ + TILE_H - 1) / TILE_H,   // 65
              B_);                           // 8
    hipLaunchKernelGGL(MarchingSquares_63058709840404_kernel, grid, block, 0,
                       stream, coarse, out);
}